// VGAE_tfp2_23356032156163
// MI455X (gfx1250) — compile-verified
//
#include <hip/hip_runtime.h>

typedef __attribute__((ext_vector_type(2))) float v2f;
typedef __attribute__((ext_vector_type(8))) float v8f;

#define NN 8192
#define FF 512
#define EE 524288
#define HH 16
#define ZZ 7
#define ZP 8   // latent dim padded to 8 for WMMA K and alignment

// workspace layout (float offsets)
#define OFF_XW1 0                      // [NN*HH]
#define OFF_H1  (OFF_XW1 + NN*HH)      // [NN*HH]  spmm1 accumulator -> h1
#define OFF_T2  (OFF_H1  + NN*HH)      // [NN*ZP]  h1 @ W2
#define OFF_T3  (OFF_T2  + NN*ZP)      // [NN*ZP]  h1 @ W3
#define OFF_ZM  (OFF_T3  + NN*ZP)      // [NN*ZP]  spmm accumulator (mean)
#define OFF_ZS  (OFF_ZM  + NN*ZP)      // [NN*ZP]  spmm accumulator (log_std)
#define OFF_Z   (OFF_ZS  + NN*ZP)      // [NN*ZP]  final z, padded col 7 = 0

__global__ void fill_zero_kernel(float* __restrict__ p, int n) {
    int i = blockIdx.x * blockDim.x + threadIdx.x;
    if (i < n) p[i] = 0.0f;
}

// ---------------------------------------------------------------------------
// Kernel 1: XW1 = X @ W1 via V_WMMA_F32_16X16X4_F32.
// One wave per 16-row tile of X (512 tiles). K-loop of 128 WMMAs.
// A 16x4 f32 fragment layout: lanes 0-15 hold K={0,1}, lanes 16-31 hold K={2,3}.
// ---------------------------------------------------------------------------
__global__ void gemm_xw1_kernel(const float* __restrict__ X,
                                const float* __restrict__ W1,
                                float* __restrict__ XW1) {
    int wave = (blockIdx.x * blockDim.x + threadIdx.x) >> 5;  // 0..511
    int lane = threadIdx.x & 31;
    int half = lane >> 4;     // 0: K pair {0,1}; 1: K pair {2,3}
    int l    = lane & 15;
    int tm   = wave << 4;
    int arow = tm + l;

    v8f acc = {};
    for (int k = 0; k < FF; k += 4) {
        int kk = k + half * 2;
        v2f a = *(const v2f*)(X + arow * FF + kk);       // A[M=l or l, K=kk..kk+1]
        v2f b;
        b.x = W1[kk * HH + l];                           // B[K=kk,   N=l]
        b.y = W1[(kk + 1) * HH + l];                     // B[K=kk+1, N=l]
        acc = __builtin_amdgcn_wmma_f32_16x16x4_f32(
            false, a, false, b, (short)0, acc, false, false);
    }
#pragma unroll
    for (int r = 0; r < 8; ++r) {
        int orow = tm + half * 8 + r;                    // C/D layout: halves = M 0-7 / 8-15
        XW1[orow * HH + l] = acc[r];
    }
}

// ---------------------------------------------------------------------------
// Kernel 2: SPMM #1 — H1acc[row] += w * XW1[col], one thread per (edge, feat).
// ---------------------------------------------------------------------------
__global__ void spmm1_kernel(const int* __restrict__ erow,
                             const int* __restrict__ ecol,
                             const float* __restrict__ ew,
                             const float* __restrict__ XW1,
                             float* __restrict__ H1acc) {
    unsigned tid = blockIdx.x * blockDim.x + threadIdx.x;
    unsigned e = tid >> 4;
    int f = tid & 15;
    if (e >= EE) return;
    float v = ew[e] * XW1[ecol[e] * HH + f];
    atomicAdd(&H1acc[erow[e] * HH + f], v);
}

// ---------------------------------------------------------------------------
// Kernel 3: h1 = relu(H1acc + b1) (in place); T2 = h1@W2, T3 = h1@W3 (padded).
// One thread per node.
// ---------------------------------------------------------------------------
__global__ void heads_kernel(float* __restrict__ H1acc,
                             const float* __restrict__ b1,
                             const float* __restrict__ W2,
                             const float* __restrict__ W3,
                             float* __restrict__ T2,
                             float* __restrict__ T3) {
    int n = blockIdx.x * blockDim.x + threadIdx.x;
    if (n >= NN) return;
    float h[HH];
#pragma unroll
    for (int j = 0; j < HH; ++j) {
        float t = H1acc[n * HH + j] + b1[j];
        t = t > 0.0f ? t : 0.0f;
        h[j] = t;
        H1acc[n * HH + j] = t;
    }
#pragma unroll
    for (int c = 0; c < ZP; ++c) {
        float s2 = 0.0f, s3 = 0.0f;
        if (c < ZZ) {
#pragma unroll
            for (int j = 0; j < HH; ++j) {
                s2 += h[j] * W2[j * ZZ + c];
                s3 += h[j] * W3[j * ZZ + c];
            }
        }
        T2[n * ZP + c] = s2;
        T3[n * ZP + c] = s3;
    }
}

// ---------------------------------------------------------------------------
// Kernel 4: SPMM #2 — both heads, one thread per (edge, padded-feat).
// ---------------------------------------------------------------------------
__global__ void spmm2_kernel(const int* __restrict__ erow,
                             const int* __restrict__ ecol,
                             const float* __restrict__ ew,
                             const float* __restrict__ T2,
                             const float* __restrict__ T3,
                             float* __restrict__ ZM,
                             float* __restrict__ ZS) {
    unsigned tid = blockIdx.x * blockDim.x + threadIdx.x;
    unsigned e = tid >> 3;
    int f = tid & 7;
    if (e >= EE || f >= ZZ) return;
    int c = ecol[e], r = erow[e];
    float w = ew[e];
    atomicAdd(&ZM[r * ZP + f], w * T2[c * ZP + f]);
    atomicAdd(&ZS[r * ZP + f], w * T3[c * ZP + f]);
}

// ---------------------------------------------------------------------------
// Kernel 5: z = (ZM + b2) + (ZS + b3) * eps, stored [NN, ZP] with pad col = 0.
// ---------------------------------------------------------------------------
__global__ void finz_kernel(const float* __restrict__ ZM,
                            const float* __restrict__ ZS,
                            const float* __restrict__ b2,
                            const float* __restrict__ b3,
                            const float* __restrict__ eps,
                            float* __restrict__ Zp) {
    int tid = blockIdx.x * blockDim.x + threadIdx.x;
    if (tid >= NN * ZP) return;
    int f = tid & 7;
    int r = tid >> 3;
    float out = 0.0f;
    if (f < ZZ) {
        float zm = ZM[tid] + b2[f];
        float zs = ZS[tid] + b3[f];
        out = zm + zs * eps[r * ZZ + f];
    }
    Zp[tid] = out;
}

// ---------------------------------------------------------------------------
// Kernel 6: recon = z @ z.T, WMMA f32 16x16x4, K=8 (2 WMMAs per 16x16 tile).
// 512x512 tiles, one wave per tile. Store-bandwidth bound (256 MB out).
// ---------------------------------------------------------------------------
__global__ void recon_kernel(const float* __restrict__ Zp,
                             float* __restrict__ out) {
    int wave = (blockIdx.x * blockDim.x + threadIdx.x) >> 5;  // 0..262143
    int lane = threadIdx.x & 31;
    int half = lane >> 4;
    int l    = lane & 15;
    int tm = (wave >> 9) << 4;
    int tn = (wave & 511) << 4;
    int arow = tm + l;
    int brow = tn + l;   // B = z.T : B[k][n] = z[n][k]

    v8f acc = {};
#pragma unroll
    for (int k = 0; k < ZP; k += 4) {
        int kk = k + half * 2;
        v2f a = *(const v2f*)(Zp + arow * ZP + kk);
        v2f b = *(const v2f*)(Zp + brow * ZP + kk);
        acc = __builtin_amdgcn_wmma_f32_16x16x4_f32(
            false, a, false, b, (short)0, acc, false, false);
    }
#pragma unroll
    for (int r = 0; r < 8; ++r) {
        long long orow = tm + half * 8 + r;
        out[orow * (long long)NN + tn + l] = acc[r];
    }
}

extern "C" void kernel_launch(void* const* d_in, const int* in_sizes, int n_in,
                              void* d_out, int out_size, void* d_ws, size_t ws_size,
                              hipStream_t stream) {
    const float* X    = (const float*)d_in[0];
    const int*   erow = (const int*)  d_in[1];
    const int*   ecol = (const int*)  d_in[2];
    const float* ew   = (const float*)d_in[3];
    const float* eps  = (const float*)d_in[4];
    const float* W1   = (const float*)d_in[5];
    const float* b1   = (const float*)d_in[6];
    const float* W2   = (const float*)d_in[7];
    const float* b2   = (const float*)d_in[8];
    const float* W3   = (const float*)d_in[9];
    const float* b3   = (const float*)d_in[10];

    float* ws    = (float*)d_ws;
    float* XW1   = ws + OFF_XW1;
    float* H1acc = ws + OFF_H1;
    float* T2    = ws + OFF_T2;
    float* T3    = ws + OFF_T3;
    float* ZM    = ws + OFF_ZM;
    float* ZS    = ws + OFF_ZS;
    float* Zp    = ws + OFF_Z;
    float* outp  = (float*)d_out;

    // zero the atomic accumulators (H1acc; ZM+ZS are contiguous)
    fill_zero_kernel<<<(NN * HH + 255) / 256, 256, 0, stream>>>(H1acc, NN * HH);
    fill_zero_kernel<<<(2 * NN * ZP + 255) / 256, 256, 0, stream>>>(ZM, 2 * NN * ZP);

    // conv1 GEMM: 512 waves, 8 waves/block
    gemm_xw1_kernel<<<512 / 8, 256, 0, stream>>>(X, W1, XW1);

    // SPMM #1: E*16 threads
    spmm1_kernel<<<(EE * 16) / 256, 256, 0, stream>>>(erow, ecol, ew, XW1, H1acc);

    // relu + bias + dense heads: one thread per node
    heads_kernel<<<(NN + 255) / 256, 256, 0, stream>>>(H1acc, b1, W2, W3, T2, T3);

    // SPMM #2: E*8 threads (both heads per thread)
    spmm2_kernel<<<(EE * 8) / 256, 256, 0, stream>>>(erow, ecol, ew, T2, T3, ZM, ZS);

    // finalize z
    finz_kernel<<<(NN * ZP) / 256, 256, 0, stream>>>(ZM, ZS, b2, b3, eps, Zp);

    // recon = z z^T : 512*512 waves, 8 waves/block
    recon_kernel<<<(512 * 512) / 8, 256, 0, stream>>>(Zp, outp);
}